// EntireGNN_37185826849343
// MI455X (gfx1250) — compile-verified
//
#include <hip/hip_runtime.h>
#include <hip/hip_bf16.h>

#define GB   16
#define GN   10000
#define GE   160000
#define GS   128
#define GH   64
#define GEMB 64
#define GP   16

typedef __attribute__((ext_vector_type(2))) float v2f;
typedef __attribute__((ext_vector_type(4))) float v4f;
typedef __attribute__((ext_vector_type(8))) float v8f;

// ---------------- utility: zero a float buffer (float4-vectorized) ----------------
__global__ __launch_bounds__(256) void zero4_kernel(v4f* __restrict__ p, int n4) {
    int i = blockIdx.x * blockDim.x + threadIdx.x;
    if (i < n4) p[i] = (v4f){0.f, 0.f, 0.f, 0.f};
}

// ---------------- degree accumulation: deg[b, col[e]] += w[b,e] ----------------
__global__ __launch_bounds__(256) void deg_kernel(const int* __restrict__ ei,
                                                  const float* __restrict__ ew,
                                                  float* __restrict__ deg) {
    int idx = blockIdx.x * blockDim.x + threadIdx.x;   // over B*E
    if (idx >= GB * GE) return;
    int b = idx / GE;
    int e = idx - b * GE;
    int c = ei[(size_t)b * 2 * GE + GE + e];           // target node
    atomicAdd(&deg[(size_t)b * GN + c], ew[idx]);
}

// ---------------- dinv = rsqrt(deg + 1)  (self-loop weight 1 folded in) ----------------
__global__ __launch_bounds__(256) void dinv_kernel(float* __restrict__ deg) {
    int i = blockIdx.x * blockDim.x + threadIdx.x;     // over B*N
    if (i < GB * GN) {
        deg[i] = rsqrtf(deg[i] + 1.0f);                // always > 0
    }
}

// ---------------- per-edge norm (layer-invariant): nrm[e] = dinv[r]*w*dinv[c] ----------------
__global__ __launch_bounds__(256) void norm_kernel(const int* __restrict__ ei,
                                                   const float* __restrict__ ew,
                                                   const float* __restrict__ dinv,
                                                   float* __restrict__ nrm) {
    int idx = blockIdx.x * blockDim.x + threadIdx.x;   // over B*E
    if (idx >= GB * GE) return;
    int b = idx / GE;
    int e = idx - b * GE;
    int r = ei[(size_t)b * 2 * GE + e];
    int c = ei[(size_t)b * 2 * GE + GE + e];
    nrm[idx] = dinv[(size_t)b * GN + r] * ew[idx] * dinv[(size_t)b * GN + c];
}

// ---------------- fp32 WMMA GEMM: D[b, m, n] = sum_k H[b, m, k] * W[n, k] ----------------
// Hout fixed = 64. Block = 128 threads = 4 waves; wave w owns N-tile n0 = 16*w.
// grid: x = N/16 m-tiles, z = batch. W (<=64x128 fp32 = 32KB) staged in LDS.
template <int HIN>
__global__ __launch_bounds__(128) void gemm_wmma_kernel(const float* __restrict__ Hsrc,
                                                        const float* __restrict__ W,
                                                        float* __restrict__ Dst) {
    __shared__ float wlds[GH * HIN];
    const int tid = threadIdx.x;
    for (int i = tid; i < GH * HIN; i += 128) wlds[i] = W[i];
    __syncthreads();

    const int wave = tid >> 5;
    const int lane = tid & 31;
    const int half = lane >> 4;        // 0: lanes 0-15, 1: lanes 16-31
    const int lm   = lane & 15;
    const int b    = blockIdx.z;
    const int m0   = blockIdx.x * 16;
    const int n0   = wave * 16;

    const float* hrow = Hsrc + ((size_t)b * GN + m0 + lm) * HIN;  // A row for this lane
    const float* wrow = wlds + (size_t)(n0 + lm) * HIN;           // B "column" = W row

    v8f acc = {};
#pragma unroll
    for (int k0 = 0; k0 < HIN; k0 += 4) {
        const int ka = k0 + half * 2;
        // A fragment: 16x4 f32 -> 2 VGPRs: lane holds M=lm, K = 2*half + {0,1}
        v2f a = *(const v2f*)(hrow + ka);
        // B fragment: 4x16 f32 -> 2 VGPRs: lane holds N=lm, K = 2*half + {0,1}
        v2f bf = *(const v2f*)(wrow + ka);
        acc = __builtin_amdgcn_wmma_f32_16x16x4_f32(
            /*neg_a=*/false, a, /*neg_b=*/false, bf,
            /*c_mod=*/(short)0, acc, /*reuse_a=*/false, /*reuse_b=*/false);
    }

    // C/D layout: VGPR j -> row (j + 8*half), col lm
    float* drow = Dst + ((size_t)b * GN + m0) * GH + n0 + lm;
#pragma unroll
    for (int j = 0; j < 8; ++j) {
        drow[(size_t)(j + 8 * half) * GH] = acc[j];
    }
}

// ---------------- edge scatter: agg[b, col, k:k+4] += tmp[b, row, k:k+4] * nrm[e] ----------------
// 16 threads per edge, each handling a float4 feature chunk (coalesced 16B loads).
__global__ __launch_bounds__(256) void scatter_kernel(const int* __restrict__ ei,
                                                      const float* __restrict__ nrm,
                                                      const float* __restrict__ tmp,
                                                      float* __restrict__ agg) {
    long long idx = (long long)blockIdx.x * blockDim.x + threadIdx.x; // over B*E*16
    if (idx >= (long long)GB * GE * (GH / 4)) return;
    int k4 = (int)(idx & 15);                  // which float4 chunk
    long long eg = idx >> 4;                   // b*E + e
    int b = (int)(eg / GE);
    int e = (int)(eg - (long long)b * GE);
    int r = ei[(size_t)b * 2 * GE + e];        // source
    int c = ei[(size_t)b * 2 * GE + GE + e];   // target
    float s = nrm[eg];
    v4f v = *(const v4f*)(tmp + ((size_t)b * GN + r) * GH + k4 * 4) * s;
    float* dst = agg + ((size_t)b * GN + c) * GH + k4 * 4;
    atomicAdd(dst + 0, v.x);
    atomicAdd(dst + 1, v.y);
    atomicAdd(dst + 2, v.z);
    atomicAdd(dst + 3, v.w);
}

// ---------------- fused self-loop + bias + optional tanh (float4) ----------------
// h_out[b,i,k] = act( agg[b,i,k] + tmp[b,i,k]*dinv[b,i]^2 + bias[k] )
__global__ __launch_bounds__(256) void act_kernel(const float* __restrict__ agg,
                                                  const float* __restrict__ tmp,
                                                  const float* __restrict__ dinv,
                                                  const float* __restrict__ bias,
                                                  float* __restrict__ hout,
                                                  int do_tanh) {
    int idx = blockIdx.x * blockDim.x + threadIdx.x;   // over B*N*16 float4 chunks
    if (idx >= GB * GN * (GH / 4)) return;
    int k4  = idx & 15;
    int node = idx >> 4;                               // b*N + i
    float di = dinv[node];
    float sl = di * di;
    v4f a = *(const v4f*)(agg + (size_t)idx * 4);
    v4f t = *(const v4f*)(tmp + (size_t)idx * 4);
    v4f bb = *(const v4f*)(bias + k4 * 4);
    v4f v = a + t * sl + bb;
    if (do_tanh) {
        v.x = tanhf(v.x); v.y = tanhf(v.y); v.z = tanhf(v.z); v.w = tanhf(v.w);
    }
    *(v4f*)(hout + (size_t)idx * 4) = v;
}

// ---------------- final gather of pos rows ----------------
__global__ __launch_bounds__(256) void gather_kernel(const float* __restrict__ emb,
                                                     const int* __restrict__ pos,
                                                     float* __restrict__ out) {
    int idx = blockIdx.x * blockDim.x + threadIdx.x;   // over B*P*EMB
    if (idx >= GB * GP * GEMB) return;
    int k = idx & (GEMB - 1);
    int bp = idx >> 6;                                 // b*P + p
    int b = bp / GP;
    int pv = pos[bp];
    out[idx] = (pv >= 0) ? emb[((size_t)b * GN + pv) * GH + k] : -3.0f;
}

extern "C" void kernel_launch(void* const* d_in, const int* in_sizes, int n_in,
                              void* d_out, int out_size, void* d_ws, size_t ws_size,
                              hipStream_t stream) {
    const float* x   = (const float*)d_in[0];  // [B, N, S]
    const int*   ei  = (const int*)d_in[1];    // [B, 2, E]
    const float* ew  = (const float*)d_in[2];  // [B, E]
    const int*   pos = (const int*)d_in[3];    // [B, P]
    const float* W1  = (const float*)d_in[4];  // [H, S]
    const float* b1  = (const float*)d_in[5];
    const float* W2  = (const float*)d_in[6];  // [H, H]
    const float* b2  = (const float*)d_in[7];
    const float* W3  = (const float*)d_in[8];  // [EMB, H]
    const float* b3  = (const float*)d_in[9];
    float* out = (float*)d_out;

    // workspace layout (floats)
    float* ws   = (float*)d_ws;
    const size_t nBN  = (size_t)GB * GN;        // 160,000
    const size_t nBE  = (size_t)GB * GE;        // 2,560,000
    const size_t nBNH = (size_t)GB * GN * GH;   // 10,240,000
    float* dinv = ws;                 // [B*N]
    float* nrm  = dinv + nBN;         // [B*E]    per-edge norm (layer-invariant)
    float* tmp  = nrm + nBE;          // [B*N*H]  GEMM output
    float* agg  = tmp + nBNH;         // [B*N*H]  scatter accumulator
    float* hbuf = agg + nBNH;         // [B*N*H]  layer activations

    const int T = 256;
    dim3 gemm_grid(GN / 16, 1, GB);   // 625 m-tiles x 16 graphs, 4 waves = 4 n-tiles

    // --- degrees & normalization (computed once, reused by all 3 layers) ---
    zero4_kernel<<<((int)(nBN / 4) + T - 1) / T, T, 0, stream>>>((v4f*)dinv, (int)(nBN / 4));
    deg_kernel<<<(GB * GE + T - 1) / T, T, 0, stream>>>(ei, ew, dinv);
    dinv_kernel<<<((int)nBN + T - 1) / T, T, 0, stream>>>(dinv);
    norm_kernel<<<(GB * GE + T - 1) / T, T, 0, stream>>>(ei, ew, dinv, nrm);

    const long long scatterN = (long long)GB * GE * (GH / 4);
    const int scatterBlocks = (int)((scatterN + T - 1) / T);
    const int elem4Blocks = (int)((nBNH / 4 + T - 1) / T);

    // --- layer 1: x[S=128] -> h[64] ---
    gemm_wmma_kernel<GS><<<gemm_grid, 128, 0, stream>>>(x, W1, tmp);
    zero4_kernel<<<elem4Blocks, T, 0, stream>>>((v4f*)agg, (int)(nBNH / 4));
    scatter_kernel<<<scatterBlocks, T, 0, stream>>>(ei, nrm, tmp, agg);
    act_kernel<<<elem4Blocks, T, 0, stream>>>(agg, tmp, dinv, b1, hbuf, 1);

    // --- layer 2: h[64] -> h[64] ---
    gemm_wmma_kernel<GH><<<gemm_grid, 128, 0, stream>>>(hbuf, W2, tmp);
    zero4_kernel<<<elem4Blocks, T, 0, stream>>>((v4f*)agg, (int)(nBNH / 4));
    scatter_kernel<<<scatterBlocks, T, 0, stream>>>(ei, nrm, tmp, agg);
    act_kernel<<<elem4Blocks, T, 0, stream>>>(agg, tmp, dinv, b2, hbuf, 1);

    // --- layer 3: h[64] -> emb[64] (no tanh) ---
    gemm_wmma_kernel<GH><<<gemm_grid, 128, 0, stream>>>(hbuf, W3, tmp);
    zero4_kernel<<<elem4Blocks, T, 0, stream>>>((v4f*)agg, (int)(nBNH / 4));
    scatter_kernel<<<scatterBlocks, T, 0, stream>>>(ei, nrm, tmp, agg);
    act_kernel<<<elem4Blocks, T, 0, stream>>>(agg, tmp, dinv, b3, hbuf, 0);

    // --- gather pos rows ---
    gather_kernel<<<(GB * GP * GEMB + T - 1) / T, T, 0, stream>>>(hbuf, pos, out);
}